// Unpool_1236950581490
// MI455X (gfx1250) — compile-verified
//
#include <hip/hip_runtime.h>

// Unpool (conv_transpose2d, stride 2, kernel has single 1 at (0,0)):
//   out[b,c,2h,2w] = x[b,c,h,w]; everything else = 0.
// Pure streaming scatter: ~103 MB read + ~411 MB write -> ~22 us floor at
// 23.3 TB/s. Strategy: one thread per input float4; two NT b128 stores of
// interleaved {x,0,x,0} on the even output row, and two async
// global_store_async_from_lds_b128 of an LDS-resident zero tile on the odd
// output row (CDNA5 async path, ASYNCcnt-tracked). NT hints keep the 514 MB
// one-shot stream out of the 192 MB L2.

typedef float v4f __attribute__((ext_vector_type(4)));

#define IN_W   112
#define IN_H   112
#define W4     (IN_W / 4)        // 28 float4 per input row
#define OUT_W  (IN_W * 2)        // 224
#define OUT_PLANE (OUT_W * IN_H * 2)  // 224*224 = 50176

__global__ __launch_bounds__(256) void unpool2x2_kernel(
    const float* __restrict__ x, float* __restrict__ out, int n4) {

    // 16B of zeros in LDS, shared (broadcast read) by all lanes' async stores.
    __shared__ v4f ztile;
    if (threadIdx.x == 0) {
        v4f z = {0.0f, 0.0f, 0.0f, 0.0f};
        ztile = z;
    }
    __syncthreads();   // DS write visible before async engine reads LDS

    // LDS byte offset of the zero tile (AS3 pointer value == LDS address).
    unsigned lds_zero =
        (unsigned)(unsigned long long)(__attribute__((address_space(3))) v4f*)&ztile;

    int idx4 = blockIdx.x * blockDim.x + threadIdx.x;
    if (idx4 < n4) {
        int w4 = idx4 % W4;
        int t  = idx4 / W4;
        int h  = t % IN_H;
        int bc = t / IN_H;

        // Streaming, read-once input: non-temporal 128-bit load.
        v4f v = __builtin_nontemporal_load((const v4f*)x + idx4);

        v4f lo = {v.x, 0.0f, v.y, 0.0f};
        v4f hi = {v.z, 0.0f, v.w, 0.0f};

        size_t obase = (size_t)bc * OUT_PLANE + (size_t)(2 * h) * OUT_W
                     + (size_t)(8 * w4);
        float* row0 = out + obase;        // even row: interleaved data/zeros
        float* row1 = row0 + OUT_W;       // odd row: all zeros

        // Even row: two NT b128 stores (32 B contiguous per thread).
        __builtin_nontemporal_store(lo, (v4f*)row0);
        __builtin_nontemporal_store(hi, (v4f*)(row0 + 4));

        // Odd row: CDNA5 async store from LDS (zeros), NT temporal hint.
        unsigned long long g1 = (unsigned long long)row1;
        asm volatile(
            "global_store_async_from_lds_b128 %0, %1, off th:TH_STORE_NT\n\t"
            "global_store_async_from_lds_b128 %0, %1, off offset:16 th:TH_STORE_NT"
            :
            : "v"(g1), "v"(lds_zero)
            : "memory");
    }

    // Drain ASYNCcnt before wave exit (s_endpgm also implies wait-idle).
    asm volatile("s_wait_asynccnt 0" ::: "memory");
}

extern "C" void kernel_launch(void* const* d_in, const int* in_sizes, int n_in,
                              void* d_out, int out_size, void* d_ws, size_t ws_size,
                              hipStream_t stream) {
    const float* x = (const float*)d_in[0];
    float* out = (float*)d_out;

    int n4 = in_sizes[0] / 4;               // 6,422,528 float4s
    int threads = 256;                      // 8 wave32s per block
    int blocks = (n4 + threads - 1) / threads;  // exactly 25,088
    unpool2x2_kernel<<<blocks, threads, 0, stream>>>(x, out, n4);
}